// neko_pystat_20100446945756
// MI455X (gfx1250) — compile-verified
//
#include <hip/hip_runtime.h>

// ---------------------------------------------------------------------------
// Histogram + gather/normalize for the "pystat" reference.
//   hist (u32, in d_ws)  <- scatter-add over gdict[flatten_label]
//   out[i] = max((cnts[g] + hist[g]) / (total + N), 0.01),  g = gdict[i]
//
// MI455X reasoning: 67 MB label stream @ 23.3 TB/s ~= 3 us floor; the 3.6 MB
// histogram + 2 MB gdict live in the 192 MB L2, so u32 atomics resolve in L2.
// Labels are staged through LDS with gfx1250 async-tensor copies
// (global_load_async_to_lds_b128) in a barrier-free double-buffered pipeline:
// each thread consumes only the 16 B it issued, so a per-wave
// s_wait_asynccnt is the only synchronization required.
// ---------------------------------------------------------------------------

#define HIST_BLOCK  256
#define TILE_LABELS 1024   // 256 threads x 4 labels (one b128 per thread/tile)
#define HIST_GRID   2048

__global__ void neko_zero_hist(unsigned* __restrict__ hist, int cap) {
    int i = blockIdx.x * blockDim.x + threadIdx.x;
    if (i < cap) hist[i] = 0u;
}

__global__ __launch_bounds__(HIST_BLOCK)
void neko_hist_kernel(const int* __restrict__ labels,
                      const int* __restrict__ gdict,
                      unsigned* __restrict__ hist,
                      int n, int numFullTiles) {
    __shared__ alignas(16) int smem[2][TILE_LABELS];

    const int t4 = threadIdx.x * 4;
    // LDS byte offsets of this thread's 16B staging slots (addrspace(3) ptrtoint)
    const unsigned lds0 = (unsigned)(unsigned long long)(&smem[0][t4]);
    const unsigned lds1 = (unsigned)(unsigned long long)(&smem[1][t4]);

    int tile = (int)blockIdx.x;
    int buf  = 0;

    // Prologue: stage the first tile asynchronously into buffer 0.
    if (tile < numFullTiles) {
        unsigned long long ga =
            (unsigned long long)(labels + (long)tile * TILE_LABELS + t4);
        asm volatile("global_load_async_to_lds_b128 %0, %1, off"
                     :: "v"(lds0), "v"(ga) : "memory");
    }

    while (tile < numFullTiles) {
        const int next = tile + (int)gridDim.x;
        if (next < numFullTiles) {
            // Issue next tile into the other buffer, then wait until only it
            // is outstanding (async loads complete in order -> current tile
            // is guaranteed resident).
            const unsigned ldsNext = buf ? lds0 : lds1;
            unsigned long long ga =
                (unsigned long long)(labels + (long)next * TILE_LABELS + t4);
            asm volatile("global_load_async_to_lds_b128 %0, %1, off"
                         :: "v"(ldsNext), "v"(ga) : "memory");
            asm volatile("s_wait_asynccnt 0x1" ::: "memory");
        } else {
            asm volatile("s_wait_asynccnt 0x0" ::: "memory");
        }

        // Consume the 4 labels this thread staged (no cross-thread sharing,
        // so no workgroup barrier is needed).
        const int4 v = *(const int4*)(&smem[buf][t4]);
        atomicAdd(&hist[gdict[v.x]], 1u);
        atomicAdd(&hist[gdict[v.y]], 1u);
        atomicAdd(&hist[gdict[v.z]], 1u);
        atomicAdd(&hist[gdict[v.w]], 1u);

        buf ^= 1;
        tile = next;
    }

    // Tail (n not a multiple of TILE_LABELS): direct loads, grid-stride.
    const long start  = (long)numFullTiles * TILE_LABELS;
    const long stride = (long)gridDim.x * blockDim.x;
    for (long i = start + (long)blockIdx.x * blockDim.x + threadIdx.x;
         i < (long)n; i += stride) {
        atomicAdd(&hist[gdict[labels[i]]], 1u);
    }
}

__global__ void neko_finalize_kernel(const int* __restrict__ gdict,
                                     const float* __restrict__ cnts,
                                     const unsigned* __restrict__ hist,
                                     const float* __restrict__ totalPtr,
                                     float* __restrict__ out,
                                     int llen, float nTok) {
    int i = blockIdx.x * blockDim.x + threadIdx.x;
    if (i < llen) {
        const float newTotal = totalPtr[0] + nTok;  // matches ref f32 order
        const int g = gdict[i];
        const float v = (cnts[g] + (float)hist[g]) / newTotal;
        out[i] = fmaxf(v, 0.01f);
    }
}

extern "C" void kernel_launch(void* const* d_in, const int* in_sizes, int n_in,
                              void* d_out, int out_size, void* d_ws, size_t ws_size,
                              hipStream_t stream) {
    // setup_inputs order: gdict, flatten_label, llen, cnts, total
    const int*   gdict  = (const int*)  d_in[0];
    const int*   labels = (const int*)  d_in[1];
    const float* cnts   = (const float*)d_in[3];
    const float* total  = (const float*)d_in[4];

    const int n   = in_sizes[1];   // token count
    const int cap = in_sizes[3];   // histogram capacity (MAX_CAPACITY)

    unsigned* hist = (unsigned*)d_ws;  // cap * 4 bytes of scratch

    // 1) zero histogram scratch (d_ws is poisoned, never pre-zeroed)
    neko_zero_hist<<<(cap + 255) / 256, 256, 0, stream>>>(hist, cap);

    // 2) histogram with async-LDS double-buffered label streaming
    const int numFullTiles = n / TILE_LABELS;
    neko_hist_kernel<<<HIST_GRID, HIST_BLOCK, 0, stream>>>(
        labels, gdict, hist, n, numFullTiles);

    // 3) gather counts, normalize by (total + N), clip from below at 0.01
    neko_finalize_kernel<<<(out_size + 255) / 256, 256, 0, stream>>>(
        gdict, cnts, hist, total, (float*)d_out, out_size, (float)n);
}